// GCNBlock_67173288509942
// MI455X (gfx1250) — compile-verified
//
#include <hip/hip_runtime.h>
#include <hip/hip_bf16.h>

typedef __attribute__((ext_vector_type(2))) float v2f;
typedef __attribute__((ext_vector_type(8))) float v8f;

#define SLOPE 0.1f
#define EPS   1e-5f

__device__ __forceinline__ float lrelu(float v) { return v >= 0.f ? v : SLOPE * v; }

// ---------------- column stats: stats[0..63]=sum, stats[64..127]=sumsq ----------------
__global__ __launch_bounds__(256) void stats_kernel(const float* __restrict__ X,
                                                    float* __restrict__ stats, int n) {
  __shared__ float ls[4][64];
  __shared__ float lq[4][64];
  const int col = threadIdx.x & 63;
  const int rg  = threadIdx.x >> 6;
  float s = 0.f, q = 0.f;
  for (int r = blockIdx.x * 4 + rg; r < n; r += gridDim.x * 4) {
    float v = X[(size_t)r * 64 + col];
    s += v; q += v * v;
  }
  ls[rg][col] = s; lq[rg][col] = q;
  __syncthreads();
  if (rg == 0) {
    s = ls[0][col] + ls[1][col] + ls[2][col] + ls[3][col];
    q = lq[0][col] + lq[1][col] + lq[2][col] + lq[3][col];
    atomicAdd(&stats[col], s);
    atomicAdd(&stats[64 + col], q);
  }
}

// mi[0..63]=mean, mi[64..127]=rsqrt(var+eps)
__global__ __launch_bounds__(64) void finalize_stats_kernel(const float* __restrict__ stats,
                                                            float* __restrict__ mi, float n) {
  int c = threadIdx.x;
  float mean = stats[c] / n;
  float var  = stats[64 + c] / n - mean * mean;
  mi[c]      = mean;
  mi[64 + c] = rsqrtf(var + EPS);
}

// ---------------- degree helpers ----------------
__global__ __launch_bounds__(256) void fill_kernel(float* __restrict__ p, float v, int n) {
  int i = blockIdx.x * 256 + threadIdx.x;
  if (i < n) p[i] = v;
}

__global__ __launch_bounds__(256) void edge_deg_kernel(const int* __restrict__ col,
                                                       const float* __restrict__ eattr,
                                                       float* __restrict__ deg, int E) {
  int e = blockIdx.x * 256 + threadIdx.x;
  if (e < E) atomicAdd(&deg[col[e]], eattr[(size_t)e * 2]);
}

__global__ __launch_bounds__(256) void dinv_kernel(const float* __restrict__ deg,
                                                   float* __restrict__ dinv, int n) {
  int i = blockIdx.x * 256 + threadIdx.x;
  if (i < n) {
    float d = deg[i];
    dinv[i] = d > 0.f ? rsqrtf(fmaxf(d, 1e-12f)) : 0.f;
  }
}

// ---------------- fused BN + leaky-ReLU + (16x64)x(64x64) WMMA GEMM ----------------
// One wave computes a 16-row output strip: 4 tiles of 16x16, K=64 in steps of 4,
// using V_WMMA_F32_16X16X4_F32 (exact fp32 MACC -> matches fp32 reference numerics).
__global__ __launch_bounds__(32) void bn_lrelu_gemm_kernel(
    const float* __restrict__ X, const float* __restrict__ mi,
    const float* __restrict__ gamma, const float* __restrict__ beta,
    const float* __restrict__ W, float* __restrict__ Y) {
  __shared__ float As[16][68];                 // padded to dodge LDS bank conflicts
  const int m0   = blockIdx.x * 16;
  const int lane = threadIdx.x;                // wave32

  // cooperative, coalesced load of the 16x64 A tile with BN + lrelu applied
  for (int i = lane; i < 16 * 64; i += 32) {
    int r = i >> 6, k = i & 63;
    float v = X[(size_t)(m0 + r) * 64 + k];
    v = (v - mi[k]) * mi[64 + k] * gamma[k] + beta[k];
    As[r][k] = lrelu(v);
  }
  __syncthreads();                             // single-wave WG: lowers to waits only

  const int half = lane >> 4;                  // 0: K{0,1} / 1: K{2,3} per ISA A layout
  const int l16  = lane & 15;
  v8f acc[4] = {};
#pragma unroll
  for (int k0 = 0; k0 < 64; k0 += 4) {
    // A fragment: lane<16 -> (A[l16][k0],A[l16][k0+1]); lane>=16 -> K+2
    v2f a = *(const v2f*)&As[l16][k0 + 2 * half];
    const float* wr = &W[(size_t)(k0 + 2 * half) * 64 + l16];
#pragma unroll
    for (int nt = 0; nt < 4; ++nt) {
      // B fragment (4x16 tile): b.x = W[k][n], b.y = W[k+1][n]
      v2f b;
      b.x = wr[nt * 16];
      b.y = wr[64 + nt * 16];
      acc[nt] = __builtin_amdgcn_wmma_f32_16x16x4_f32(
          /*neg_a=*/false, a, /*neg_b=*/false, b,
          /*c_mod=*/(short)0, acc[nt], /*reuse_a=*/false, /*reuse_b=*/false);
    }
  }

  // C/D layout: VGPR v, lanes 0-15 -> M=v, lanes 16-31 -> M=8+v; N = lane%16
#pragma unroll
  for (int nt = 0; nt < 4; ++nt)
#pragma unroll
    for (int v = 0; v < 8; ++v)
      Y[(size_t)(m0 + v + 8 * half) * 64 + nt * 16 + l16] = acc[nt][v];
}

// ---------------- edge scatter: agg[c] += H[r] * dinv[r]*w*dinv[c] ----------------
__global__ __launch_bounds__(256) void scatter_kernel(
    const float* __restrict__ H, const int* __restrict__ row, const int* __restrict__ col,
    const float* __restrict__ eattr, const float* __restrict__ dinv,
    float* __restrict__ agg, long long E) {
  long long idx = (long long)blockIdx.x * 256 + threadIdx.x;
  if (idx < E * 64) {
    long long e = idx >> 6;                    // wave-uniform -> scalar broadcast loads
    int f = (int)(idx & 63);
    int r = row[e], c = col[e];
    float norm = dinv[r] * eattr[(size_t)e * 2] * dinv[c];
    atomicAdd(&agg[(size_t)c * 64 + f], H[(size_t)r * 64 + f] * norm);
  }
}

// ---------------- self-loop + bias + lrelu: out = lrelu(agg + H*dinv^2 + b) ----------------
__global__ __launch_bounds__(256) void post_agg_kernel(
    const float* __restrict__ agg, const float* __restrict__ H,
    const float* __restrict__ dinv, const float* __restrict__ bias,
    float* __restrict__ out, int n) {
  int idx = blockIdx.x * 256 + threadIdx.x;
  if (idx < n * 64) {
    int i = idx >> 6, f = idx & 63;
    float di = dinv[i];
    float v = agg[idx] + H[idx] * di * di + bias[f];
    out[idx] = lrelu(v);
  }
}

// ---------------- edge_index passthrough (second tuple output), int -> float ----------------
__global__ __launch_bounds__(256) void copy_edges_kernel(const int* __restrict__ ei,
                                                         float* __restrict__ out, int total) {
  int i = blockIdx.x * 256 + threadIdx.x;
  if (i < total) out[i] = (float)ei[i];
}

extern "C" void kernel_launch(void* const* d_in, const int* in_sizes, int n_in,
                              void* d_out, int out_size, void* d_ws, size_t ws_size,
                              hipStream_t stream) {
  const float* x     = (const float*)d_in[0];
  const int*   eidx  = (const int*)d_in[1];
  const float* eattr = (const float*)d_in[2];
  const float* bn1_g = (const float*)d_in[3];
  const float* bn1_b = (const float*)d_in[4];
  const float* W1    = (const float*)d_in[5];
  const float* b1    = (const float*)d_in[6];
  const float* bn2_g = (const float*)d_in[7];
  const float* bn2_b = (const float*)d_in[8];
  const float* W2    = (const float*)d_in[9];
  const float* b2    = (const float*)d_in[10];
  float* out = (float*)d_out;

  const int N = in_sizes[0] / 64;
  const int E = in_sizes[1] / 2;
  const int* row = eidx;
  const int* col = eidx + E;

  // workspace layout (floats)
  float* ws    = (float*)d_ws;
  float* stats = ws;                                   // 128
  float* mi1   = ws + 128;                             // 128
  float* mi2   = ws + 256;                             // 128
  float* deg   = ws + 384;                             // N
  float* dinv  = deg + N;                              // N
  float* hw    = dinv + N;                             // N*64
  float* agg   = hw + (size_t)N * 64;                  // N*64
  float* h2    = agg + (size_t)N * 64;                 // N*64

  const int nb256   = (N + 255) / 256;
  const int eb256   = (E + 255) / 256;
  const int featb   = (N * 64 + 255) / 256;
  const unsigned sb = (unsigned)(((long long)E * 64 + 255) / 256);

  // degree / norm (recomputed every call: no state carried across calls)
  fill_kernel<<<nb256, 256, 0, stream>>>(deg, 1.0f, N);               // self-loop weight
  edge_deg_kernel<<<eb256, 256, 0, stream>>>(col, eattr, deg, E);
  dinv_kernel<<<nb256, 256, 0, stream>>>(deg, dinv, N);

  // ---- layer 1: lrelu(BN(x)) @ W1 -> scatter -> +b1, lrelu ----
  hipMemsetAsync(stats, 0, 128 * sizeof(float), stream);
  stats_kernel<<<1024, 256, 0, stream>>>(x, stats, N);
  finalize_stats_kernel<<<1, 64, 0, stream>>>(stats, mi1, (float)N);
  bn_lrelu_gemm_kernel<<<N / 16, 32, 0, stream>>>(x, mi1, bn1_g, bn1_b, W1, hw);
  hipMemsetAsync(agg, 0, (size_t)N * 64 * sizeof(float), stream);
  scatter_kernel<<<sb, 256, 0, stream>>>(hw, row, col, eattr, dinv, agg, (long long)E);
  post_agg_kernel<<<featb, 256, 0, stream>>>(agg, hw, dinv, b1, h2, N);

  // ---- layer 2: lrelu(BN(h2)) @ W2 -> scatter -> +b2, lrelu -> out ----
  hipMemsetAsync(stats, 0, 128 * sizeof(float), stream);
  stats_kernel<<<1024, 256, 0, stream>>>(h2, stats, N);
  finalize_stats_kernel<<<1, 64, 0, stream>>>(stats, mi2, (float)N);
  bn_lrelu_gemm_kernel<<<N / 16, 32, 0, stream>>>(h2, mi2, bn2_g, bn2_b, W2, hw);
  hipMemsetAsync(agg, 0, (size_t)N * 64 * sizeof(float), stream);
  scatter_kernel<<<sb, 256, 0, stream>>>(hw, row, col, eattr, dinv, agg, (long long)E);
  post_agg_kernel<<<featb, 256, 0, stream>>>(agg, hw, dinv, b2, out, N);

  // ---- second tuple output: edge_index ----
  copy_edges_kernel<<<(2 * E + 255) / 256, 256, 0, stream>>>(eidx, out + (size_t)N * 64, 2 * E);
}